// FCOSDetector_76038101008472
// MI455X (gfx1250) — compile-verified
//
#include <hip/hip_runtime.h>
#include <stdint.h>

// ---------------- problem constants ----------------
#define NPOS      21824      // total anchor positions per image
#define NBATCH    8
#define NCLS      80
#define MDET      1000
#define MPAD      1024
#define CAP       6144       // candidate cap for exact rank-select
#define SCORE_THR 0.05f
#define IOU_THR   0.6f
#define NCHUNK    171        // 128+32+8+2+1 position-chunks of 128
#define CHUNK     128

__device__ constexpr int LVL_A[5]  = {16384, 4096, 1024, 256, 64};
__device__ constexpr int LVL_W[5]  = {128, 64, 32, 16, 8};
__device__ constexpr int LVL_S[5]  = {8, 16, 32, 64, 128};
__device__ constexpr int LVL_B0[5] = {0, 16384, 20480, 21504, 21760};

struct LvlPtrs { const float* cls[5]; const float* cnt[5]; const float* reg[5]; };

typedef unsigned int v4u __attribute__((ext_vector_type(4)));
typedef int          v4i __attribute__((ext_vector_type(4)));
typedef int          v8i __attribute__((ext_vector_type(8)));
typedef float        v2f __attribute__((ext_vector_type(2)));
typedef float        v8f __attribute__((ext_vector_type(8)));

__device__ __forceinline__ float sigmoidf(float x) { return 1.0f / (1.0f + __expf(-x)); }

__device__ __forceinline__ int lvl_from_chunk(int c, int& ci) {
  if (c < 128) { ci = c;       return 0; }
  if (c < 160) { ci = c - 128; return 1; }
  if (c < 168) { ci = c - 160; return 2; }
  if (c < 170) { ci = c - 168; return 3; }
  ci = c - 170; return 4;
}

// ---------------- init: zero histogram/counters/topk/mask ----------------
__global__ void k_init(unsigned* z, size_t nwords) {
  size_t i = (size_t)blockIdx.x * blockDim.x + threadIdx.x;
  size_t st = (size_t)gridDim.x * blockDim.x;
  for (; i < nwords; i += st) z[i] = 0u;
}

// ---------------- stage 1: decode (TDM tile -> LDS -> reduce) ----------------
__global__ void k_decode(LvlPtrs P, float* scores, int* classes, float* boxes,
                         unsigned* hist) {
  __shared__ float tile[NCLS * CHUNK];                 // 40 KB: 80 channels x 128 positions
  const int blk = blockIdx.x;
  const int b = blk / NCHUNK;
  int ci; const int l = lvl_from_chunk(blk % NCHUNK, ci);
  const int A = LVL_A[l], W = LVL_W[l], ST = LVL_S[l], BASE = LVL_B0[l];
  const int tileW = (A < CHUNK) ? A : CHUNK;
  const int s0 = ci * CHUNK;

#if __has_builtin(__builtin_amdgcn_tensor_load_to_lds)
  if (threadIdx.x < 32u) {                             // one wave issues the TDM
    const uint64_t ga = (uint64_t)(uintptr_t)P.cls[l]
                      + ((uint64_t)b * NCLS * A + (uint64_t)s0) * 4ull;
    // Flat LDS addr low 32 bits == LDS byte offset (ISA 10.2 aperture mapping).
    const uint32_t ldsoff = (uint32_t)(uintptr_t)&tile[0];
    v4u g0;
    g0[0] = 1u;                                        // count=1, user descriptor
    g0[1] = ldsoff;                                    // lds_addr
    g0[2] = (uint32_t)ga;                              // global_addr[31:0]
    g0[3] = ((uint32_t)(ga >> 32) & 0x1FFFFFFu) | (2u << 30);  // addr[56:32] | type=2
    const uint32_t td0 = (uint32_t)(A - s0);           // remaining extent (safe either way)
    const uint32_t td1 = NCLS;
    const uint64_t st0 = (uint64_t)A;                  // row stride in elements
    v8i g1;
    g1[0] = (int)(2u << 16);                           // wg_mask=0, data_size=4B, no iter/pad
    g1[1] = (int)((td0 & 0xFFFFu) << 16);              // tensor_dim0[15:0] @ bits 63:48
    g1[2] = (int)(((td0 >> 16) & 0xFFFFu) | ((td1 & 0xFFFFu) << 16));
    g1[3] = (int)(((td1 >> 16) & 0xFFFFu) | ((uint32_t)tileW << 16));   // tile_dim0
    g1[4] = (int)(uint32_t)NCLS;                       // tile_dim1=80, tile_dim2=0
    g1[5] = (int)(uint32_t)(st0 & 0xFFFFFFFFull);      // tensor_dim0_stride[31:0]
    g1[6] = (int)(uint32_t)((st0 >> 32) & 0xFFFFull);  // stride[47:32] | dim1_stride lo16=0
    g1[7] = 0;
    v4i z4 = {0, 0, 0, 0};
#if defined(__clang_major__) && __clang_major__ >= 23
    v8i z8 = {0, 0, 0, 0, 0, 0, 0, 0};
    __builtin_amdgcn_tensor_load_to_lds(g0, g1, z4, z4, z8, 0);
#else
    __builtin_amdgcn_tensor_load_to_lds(g0, g1, z4, z4, 0);
#endif
#if __has_builtin(__builtin_amdgcn_s_wait_tensorcnt)
    __builtin_amdgcn_s_wait_tensorcnt(0);
#else
    asm volatile("s_wait_tensorcnt 0x0" ::: "memory");
#endif
  }
#else
  // Fallback: cooperative fill of the LDS tile with plain global loads.
  for (int e = threadIdx.x; e < NCLS * tileW; e += blockDim.x) {
    const int cc = e / tileW, p = e % tileW;
    tile[e] = P.cls[l][(size_t)b * NCLS * A + (size_t)cc * A + (s0 + p)];
  }
#endif
  __syncthreads();

  const int t = threadIdx.x;
  if (t < tileW) {
    const int s = s0 + t;
    // sigmoid is monotonic: max/argmax on logits, sigmoid once.
    float best = tile[t];
    int bc = 0;
    #pragma unroll 4
    for (int cc = 1; cc < NCLS; ++cc) {
      const float v = tile[cc * tileW + t];
      if (v > best) { best = v; bc = cc; }
    }
    const float cv = P.cnt[l][(size_t)b * A + s];
    const float score = sqrtf(sigmoidf(best) * sigmoidf(cv));
    const int g = BASE + s;
    const size_t gi = (size_t)b * NPOS + g;
    scores[gi] = score;
    classes[gi] = bc + 1;
    const float lf = P.reg[l][((size_t)b * 4 + 0) * A + s];
    const float tp = P.reg[l][((size_t)b * 4 + 1) * A + s];
    const float rt = P.reg[l][((size_t)b * 4 + 2) * A + s];
    const float bt = P.reg[l][((size_t)b * 4 + 3) * A + s];
    const float x = (float)((s % W) * ST + ST / 2);
    const float y = (float)((s / W) * ST + ST / 2);
    float* bx = boxes + gi * 4;
    bx[0] = x - lf; bx[1] = y - tp; bx[2] = x + rt; bx[3] = y + bt;
    unsigned bin = __float_as_uint(score) >> 19;       // exp + 4 mantissa bits
    if (bin > 4095u) bin = 4095u;
    atomicAdd(&hist[b * 4096 + bin], 1u);
  }
}

// ---------------- stage 2: histogram threshold ----------------
__global__ void k_thresh(const unsigned* hist, unsigned* thr) {
  const int b = threadIdx.x;
  if (b < NBATCH) {
    unsigned cum = 0; int bin = 4095;
    for (; bin > 0; --bin) { cum += hist[b * 4096 + bin]; if (cum >= MDET) break; }
    thr[b] = (unsigned)bin;
  }
}

// ---------------- stage 3: compact candidates >= threshold bin ----------------
__global__ void k_compact(const float* scores, const unsigned* thr, unsigned* cnt,
                          unsigned* ckey, int* cidx) {
  const int idx = blockIdx.x * blockDim.x + threadIdx.x;
  if (idx >= NBATCH * NPOS) return;
  const int b = idx / NPOS, g = idx % NPOS;
  const unsigned bits = __float_as_uint(scores[idx]); // scores >= 0: uint-orderable
  unsigned bin = bits >> 19; if (bin > 4095u) bin = 4095u;
  if (bin >= thr[b]) {
    const unsigned pos = atomicAdd(&cnt[b], 1u);
    if (pos < CAP) { ckey[b * CAP + pos] = bits; cidx[b * CAP + pos] = g; }
  }
}

// ---------------- stage 4: exact rank-select + gather topk ----------------
__global__ void k_rank(const unsigned* ckey, const int* cidx, const unsigned* cnt,
                       const int* classes, const float* boxes,
                       float* tsc, int* tcls, float* tarea, float* tbox) {
  __shared__ unsigned sk[CAP];
  __shared__ int si[CAP];
  const int b = blockIdx.x;
  const int C = (int)min(cnt[b], (unsigned)CAP);
  for (int e = threadIdx.x; e < C; e += blockDim.x) {
    sk[e] = ckey[b * CAP + e]; si[e] = cidx[b * CAP + e];
  }
  __syncthreads();
  for (int e = threadIdx.x; e < C; e += blockDim.x) {
    const unsigned ke = sk[e]; const int ie = si[e];
    int r = 0;
    for (int j = 0; j < C; ++j) {                      // jax top_k tie-break: lower idx first
      const unsigned kj = sk[j];
      r += (int)((kj > ke) || ((kj == ke) && (si[j] < ie)));
    }
    if (r < MPAD) {
      const size_t gi = (size_t)b * NPOS + ie;
      const int o = b * MPAD + r;
      tsc[o]  = __uint_as_float(ke);
      tcls[o] = classes[gi];
      const float x1 = boxes[gi * 4 + 0], y1 = boxes[gi * 4 + 1];
      const float x2 = boxes[gi * 4 + 2], y2 = boxes[gi * 4 + 3];
      tbox[o * 4 + 0] = x1; tbox[o * 4 + 1] = y1;
      tbox[o * 4 + 2] = x2; tbox[o * 4 + 3] = y2;
      tarea[o] = (x2 - x1 + 1.0f) * (y2 - y1 + 1.0f);
    }
  }
}

// ---------------- stage 5: 16x16 IoU tiles -> bit-packed suppression mask ----
// area_i + area_j outer-sum done with one V_WMMA_F32_16X16X4_F32:
//   A(16x4): col0 = areas_row, col1 = 1      B(4x16): row0 = 1, row1 = areas_col
__global__ void k_mask(const float* tbox, const float* tarea, const int* tcls,
                       unsigned* maskw) {
  const int blk = blockIdx.x;
  const int b = blk >> 12;                             // 64*64 = 4096 tile pairs / batch
  const int r = blk & 4095;
  const int it = r >> 6, jt = r & 63;
  if (jt > it) return;                                 // only j<i bits are ever read
  __shared__ float rx1[16], ry1[16], rx2[16], ry2[16], rar[16];
  __shared__ int rcl[16];
  const int lane = threadIdx.x;
  const int i0 = it * 16, j0 = jt * 16;
  if (lane < 16) {
    const int o = b * MPAD + i0 + lane;
    rx1[lane] = tbox[o * 4 + 0]; ry1[lane] = tbox[o * 4 + 1];
    rx2[lane] = tbox[o * 4 + 2]; ry2[lane] = tbox[o * 4 + 3];
    rar[lane] = tarea[o]; rcl[lane] = tcls[o];
  }
  __syncthreads();
  const int n = lane & 15;
  const int j = j0 + n;
  const int oc = b * MPAD + j;
  const float cx1 = tbox[oc * 4 + 0], cy1 = tbox[oc * 4 + 1];
  const float cx2 = tbox[oc * 4 + 2], cy2 = tbox[oc * 4 + 3];
  const float car = tarea[oc];
  const int ccl = tcls[oc];

  v8f d;
#if __has_builtin(__builtin_amdgcn_wmma_f32_16x16x4_f32)
  v2f a, bb;
  if (lane < 16) { a.x = rar[lane]; a.y = 1.0f; bb.x = 1.0f; bb.y = car; }
  else           { a.x = 0.0f;      a.y = 0.0f; bb.x = car;  bb.y = 1.0f; }
  v8f c0 = {0.f, 0.f, 0.f, 0.f, 0.f, 0.f, 0.f, 0.f};
  d = __builtin_amdgcn_wmma_f32_16x16x4_f32(false, a, false, bb, (short)0, c0,
                                            false, false);
#else
  #pragma unroll
  for (int rr = 0; rr < 8; ++rr) d[rr] = rar[rr + ((lane >> 4) << 3)] + car;
#endif

  const unsigned wsh = (unsigned)(j0 & 16);
  const unsigned widx = (unsigned)(j0 >> 5);
  #pragma unroll
  for (int rr = 0; rr < 8; ++rr) {
    const int m = rr + ((lane >> 4) << 3);             // C/D layout: M = vgpr + 8*(lane>=16)
    const int i = i0 + m;
    const float xx1 = fmaxf(rx1[m], cx1);
    const float yy1 = fmaxf(ry1[m], cy1);
    const float xx2 = fminf(rx2[m], cx2);
    const float yy2 = fminf(ry2[m], cy2);
    const float inter = fmaxf(xx2 - xx1, 0.0f) * fmaxf(yy2 - yy1, 0.0f);
    const float iou = inter / (d[rr] - inter);         // d = area_i + area_j (WMMA)
    const bool bit = (rcl[m] == ccl) && (iou > IOU_THR) && (j < i);
    const unsigned bal = (unsigned)__ballot(bit);      // lo16: row m=rr, hi16: row m=8+rr
    if (lane == 0) {
      atomicOr(&maskw[((size_t)b * MPAD + (i0 + rr)) * 32 + widx],
               (bal & 0xFFFFu) << wsh);
      atomicOr(&maskw[((size_t)b * MPAD + (i0 + 8 + rr)) * 32 + widx],
               ((bal >> 16) & 0xFFFFu) << wsh);
    }
  }
}

// ---------------- stage 6: serial suppression, one wave32 per batch ----------
__global__ void k_nms(const float* tsc, const unsigned* maskw, unsigned* keepw) {
  const int b = blockIdx.x;
  const int lane = threadIdx.x;                        // 32 lanes, keep word per lane
  unsigned kw = 0u;
  for (int i = 0; i < MDET; ++i) {
    const unsigned m = maskw[((size_t)b * MPAD + i) * 32 + lane];
    const bool hit = (m & kw) != 0u;
    const unsigned long long bal = __ballot(hit);
    const float sc = tsc[b * MPAD + i];                // uniform scalar load
    const bool keep = (sc >= SCORE_THR) && (bal == 0ull);
    if (keep && lane == (i >> 5)) kw |= (1u << (i & 31));
  }
  keepw[b * 32 + lane] = kw;
}

// ---------------- stage 7: clip + emit ----------------
__global__ void k_out(const float* tsc, const int* tcls, const float* tbox,
                      const unsigned* keepw, float* out) {
  const int idx = blockIdx.x * blockDim.x + threadIdx.x;
  if (idx >= NBATCH * MDET) return;
  const int b = idx / MDET, i = idx % MDET;
  const int o = b * MPAD + i;
  const bool keep = (keepw[b * 32 + (i >> 5)] >> (i & 31)) & 1u;
  const float x1 = fminf(fmaxf(tbox[o * 4 + 0], 0.0f), 1023.0f);
  const float y1 = fminf(fmaxf(tbox[o * 4 + 1], 0.0f), 1023.0f);
  const float x2 = fminf(fmaxf(tbox[o * 4 + 2], 0.0f), 1023.0f);
  const float y2 = fminf(fmaxf(tbox[o * 4 + 3], 0.0f), 1023.0f);
  out[b * MDET + i] = keep ? tsc[o] : 0.0f;                       // scores
  out[NBATCH * MDET + b * MDET + i] = keep ? (float)tcls[o] : 0.0f; // classes
  float* bo = out + 2 * NBATCH * MDET + (size_t)(b * MDET + i) * 4;
  bo[0] = keep ? x1 : 0.0f; bo[1] = keep ? y1 : 0.0f;
  bo[2] = keep ? x2 : 0.0f; bo[3] = keep ? y2 : 0.0f;
}

// ---------------- host ----------------
extern "C" void kernel_launch(void* const* d_in, const int* in_sizes, int n_in,
                              void* d_out, int out_size, void* d_ws, size_t ws_size,
                              hipStream_t stream) {
  (void)in_sizes; (void)n_in; (void)out_size; (void)ws_size;
  LvlPtrs P;
  for (int l = 0; l < 5; ++l) {          // dict order: (cls, cnt, reg) per level
    P.cls[l] = (const float*)d_in[3 * l + 0];
    P.cnt[l] = (const float*)d_in[3 * l + 1];
    P.reg[l] = (const float*)d_in[3 * l + 2];
  }
  char* w = (char*)d_ws;
  size_t off = 0;
  auto alloc = [&](size_t bytes) { size_t o = off; off = (off + bytes + 255) & ~(size_t)255; return o; };

  float*    scores  = (float*)(w + alloc((size_t)NBATCH * NPOS * 4));
  int*      classes = (int*)(w + alloc((size_t)NBATCH * NPOS * 4));
  float*    boxes   = (float*)(w + alloc((size_t)NBATCH * NPOS * 16));
  unsigned* ckey    = (unsigned*)(w + alloc((size_t)NBATCH * CAP * 4));
  int*      cidx    = (int*)(w + alloc((size_t)NBATCH * CAP * 4));
  const size_t zbeg = off;               // everything below must start zeroed
  unsigned* hist    = (unsigned*)(w + alloc((size_t)NBATCH * 4096 * 4));
  unsigned* cnt     = (unsigned*)(w + alloc((size_t)NBATCH * 4));
  unsigned* thr     = (unsigned*)(w + alloc((size_t)NBATCH * 4));
  float*    tsc     = (float*)(w + alloc((size_t)NBATCH * MPAD * 4));
  int*      tcls    = (int*)(w + alloc((size_t)NBATCH * MPAD * 4));
  float*    tarea   = (float*)(w + alloc((size_t)NBATCH * MPAD * 4));
  float*    tbox    = (float*)(w + alloc((size_t)NBATCH * MPAD * 16));
  unsigned* maskw   = (unsigned*)(w + alloc((size_t)NBATCH * MPAD * 32 * 4));
  unsigned* keepw   = (unsigned*)(w + alloc((size_t)NBATCH * 32 * 4));
  const size_t zend = off;

  const size_t zwords = (zend - zbeg) / 4;
  k_init<<<512, 256, 0, stream>>>((unsigned*)(w + zbeg), zwords);
  k_decode<<<NBATCH * NCHUNK, CHUNK, 0, stream>>>(P, scores, classes, boxes, hist);
  k_thresh<<<1, 32, 0, stream>>>(hist, thr);
  k_compact<<<(NBATCH * NPOS + 255) / 256, 256, 0, stream>>>(scores, thr, cnt, ckey, cidx);
  k_rank<<<NBATCH, 256, 0, stream>>>(ckey, cidx, cnt, classes, boxes, tsc, tcls, tarea, tbox);
  k_mask<<<NBATCH * 4096, 32, 0, stream>>>(tbox, tarea, tcls, maskw);
  k_nms<<<NBATCH, 32, 0, stream>>>(tsc, maskw, keepw);
  k_out<<<(NBATCH * MDET + 255) / 256, 256, 0, stream>>>(tsc, tcls, tbox, keepw, (float*)d_out);
}